// listWiseModel_2491081031807
// MI455X (gfx1250) — compile-verified
//
#include <hip/hip_runtime.h>
#include <hip/hip_bf16.h>

typedef __attribute__((ext_vector_type(2))) float v2f;
typedef __attribute__((ext_vector_type(8))) float v8f;

#define NDOCS  300000
#define FDIM   136
#define H1DIM  128
#define H2DIM  64
#define NPAIRS 23
#define SIGMA  1.0f
#define MPAD   32   // 23 rows padded to two 16-row WMMA tiles

__launch_bounds__(256, 1)
__global__ void ltr_lambdarank_fused(
    const float* __restrict__ x,
    const float* __restrict__ W1,
    const float* __restrict__ b1,
    const float* __restrict__ W2,
    const float* __restrict__ b2,
    const float* __restrict__ Wout,
    const float* __restrict__ bout,
    const float* __restrict__ target,
    const int*   __restrict__ random_docs,
    float* __restrict__ out)
{
    __shared__ float Xs[MPAD * FDIM];    // 32x136 gathered features
    __shared__ float H1s[MPAD * H1DIM];  // 32x128
    __shared__ float H2s[MPAD * H2DIM];  // 32x64
    __shared__ int   rd[NPAIRS];
    __shared__ float lab[NPAIRS];
    __shared__ float sumL[NPAIRS], sumD[NPAIRS];
    __shared__ float cumLD[NPAIRS];
    __shared__ float sc[NPAIRS];

    const int tid  = threadIdx.x;
    const int wave = tid >> 5;
    const int lane = tid & 31;

    // ---- stage sampled doc ids + labels ----
    if (tid < NPAIRS) {
        int d = random_docs[tid];
        rd[tid]  = d;
        lab[tid] = target[d];
    }
    __syncthreads();

    // ---- per-row lambda / delta sums (23x23, trivial) ----
    if (tid < NPAIRS) {
        float li = lab[tid];
        float wi = 1.0f / (float)(tid + 1);
        int   di = rd[tid];
        float sL = 0.f, sD = 0.f;
        for (int j = 0; j < NPAIRS; ++j) {
            if (rd[j] != di) {
                float lj = lab[j];
                float wj = 1.0f / (float)(j + 1);
                float df = li - lj;
                float S  = (df > 0.f) ? 1.f : ((df < 0.f) ? -1.f : 0.f);
                sL += SIGMA * (0.5f * (1.f - S) - 0.5f);
                sD += fabsf((lj - li) * (wi - wj));
            }
        }
        sumL[tid] = sL;
        sumD[tid] = sD;
    }

    // ---- gather the 23 live feature rows into LDS (pad rows 23..31 = 0) ----
    for (int idx = tid; idx < MPAD * FDIM; idx += 256) {
        int r = idx / FDIM, c = idx - r * FDIM;
        Xs[idx] = (r < NPAIRS) ? x[(long)rd[r] * FDIM + c] : 0.f;
    }
    __syncthreads();

    // running cumulative sums (faithful to the reference's "never reset" bug)
    if (tid == 0) {
        float cL = 0.f, cD = 0.f;
        for (int i = 0; i < NPAIRS; ++i) {
            cL += sumL[i];
            cD += sumD[i];
            cumLD[i] = cL * cD;
        }
    }

    // ---- GEMM1: H1s(32x128) = relu(Xs(32x136) @ W1(136x128) + b1) ----
    // 8 waves, each owns one 16-col N-tile and both 16-row M-tiles.
    {
        const int nt    = wave;
        const int nlo   = lane & 15;
        const int khalf = (lane >> 4) << 1;     // lanes 16..31 carry K+2,K+3
        const int n     = nt * 16 + nlo;
        v8f acc0, acc1;
        float bv = b1[n];
        for (int r = 0; r < 8; ++r) { acc0[r] = bv; acc1[r] = bv; }
        for (int k0 = 0; k0 < FDIM; k0 += 4) {
            int kb = k0 + khalf;
            v2f bf, a0, a1;
            bf.x = W1[kb * H1DIM + n];
            bf.y = W1[(kb + 1) * H1DIM + n];
            a0.x = Xs[nlo * FDIM + kb];           // M-tile 0, row = lane&15
            a0.y = Xs[nlo * FDIM + kb + 1];
            a1.x = Xs[(16 + nlo) * FDIM + kb];    // M-tile 1
            a1.y = Xs[(16 + nlo) * FDIM + kb + 1];
            acc0 = __builtin_amdgcn_wmma_f32_16x16x4_f32(false, a0, false, bf,
                                                         (short)0, acc0, false, false);
            acc1 = __builtin_amdgcn_wmma_f32_16x16x4_f32(false, a1, false, bf,
                                                         (short)0, acc1, false, false);
        }
        int mbase = (lane >> 4) << 3;   // lanes 16..31 hold rows M+8 of C/D
        for (int r = 0; r < 8; ++r) {
            float v0 = acc0[r] > 0.f ? acc0[r] : 0.f;
            float v1 = acc1[r] > 0.f ? acc1[r] : 0.f;
            H1s[(mbase + r) * H1DIM + n]        = v0;
            H1s[(16 + mbase + r) * H1DIM + n]   = v1;
        }
    }
    __syncthreads();

    // ---- GEMM2: H2s(32x64) = relu(H1s(32x128) @ W2(128x64) + b2) ----
    {
        const int mt    = wave >> 2;
        const int nt    = wave & 3;
        const int nlo   = lane & 15;
        const int khalf = (lane >> 4) << 1;
        const int n     = nt * 16 + nlo;
        const int m     = mt * 16 + nlo;
        v8f acc;
        float bv = b2[n];
        for (int r = 0; r < 8; ++r) acc[r] = bv;
        for (int k0 = 0; k0 < H1DIM; k0 += 4) {
            int kb = k0 + khalf;
            v2f af, bf;
            af.x = H1s[m * H1DIM + kb];
            af.y = H1s[m * H1DIM + kb + 1];
            bf.x = W2[kb * H2DIM + n];
            bf.y = W2[(kb + 1) * H2DIM + n];
            acc = __builtin_amdgcn_wmma_f32_16x16x4_f32(false, af, false, bf,
                                                        (short)0, acc, false, false);
        }
        int mbase = mt * 16 + ((lane >> 4) << 3);
        for (int r = 0; r < 8; ++r) {
            float v = acc[r] > 0.f ? acc[r] : 0.f;
            H2s[(mbase + r) * H2DIM + n] = v;
        }
    }
    __syncthreads();

    // ---- scores (32x64 @ 64x1) + final weighted reduction ----
    if (tid < NPAIRS) {
        float s = bout[0];
        for (int k = 0; k < H2DIM; ++k) s += H2s[tid * H2DIM + k] * Wout[k];
        sc[tid] = s;
    }
    __syncthreads();
    if (tid == 0) {
        float loss = 0.f;
        for (int i = 0; i < NPAIRS; ++i) loss += cumLD[i] * sc[i];
        out[0] = loss;
    }
}

extern "C" void kernel_launch(void* const* d_in, const int* in_sizes, int n_in,
                              void* d_out, int out_size, void* d_ws, size_t ws_size,
                              hipStream_t stream) {
    (void)in_sizes; (void)n_in; (void)out_size; (void)d_ws; (void)ws_size;
    const float* x      = (const float*)d_in[0];
    const float* W1     = (const float*)d_in[1];
    const float* b1     = (const float*)d_in[2];
    const float* W2     = (const float*)d_in[3];
    const float* b2     = (const float*)d_in[4];
    const float* Wout   = (const float*)d_in[5];
    const float* bout   = (const float*)d_in[6];
    const float* target = (const float*)d_in[7];
    const int*   rdocs  = (const int*)d_in[8];
    float* out = (float*)d_out;
    ltr_lambdarank_fused<<<1, 256, 0, stream>>>(x, W1, b1, W2, b2, Wout, bout,
                                                target, rdocs, out);
}